// PolymerGCN_69286412419645
// MI455X (gfx1250) — compile-verified
//
#include <hip/hip_runtime.h>

#define N_NODES 100000
#define N_EDGES 1600000
#define N_GRAPHS 2000
#define HIDDEN 128
#define N_LAYERS 4
#define BN_EPS 1e-5f

typedef __attribute__((ext_vector_type(2))) float v2f;
typedef __attribute__((ext_vector_type(8))) float v8f;

// ---------------------------------------------------------------- utilities
__global__ void zero_f32(float* __restrict__ p, size_t n) {
    size_t i = (size_t)blockIdx.x * blockDim.x + threadIdx.x;
    if (i < n) p[i] = 0.0f;
}

// deg[dst] += 1 over all edges (deg buffer pre-zeroed)
__global__ void degree_kernel(const int* __restrict__ dst, float* __restrict__ deg) {
    int e = blockIdx.x * blockDim.x + threadIdx.x;
    if (e < N_EDGES) atomicAdd(&deg[dst[e]], 1.0f);
}

// norm[i] = rsqrt(deg[i] + 1)   (in-place over deg)
__global__ void norm_kernel(float* __restrict__ norm) {
    int i = blockIdx.x * blockDim.x + threadIdx.x;
    if (i < N_NODES) norm[i] = rsqrtf(norm[i] + 1.0f);
}

// ----------------------------------------------------- WMMA fp32 GEMM: XW = H @ W
// H: [N_NODES,128] row-major, W: [128,128] row-major, XW: [N_NODES,128].
// One wave computes a 16-row strip across all 128 columns (8 16x16 tiles),
// accumulating K=128 in 32 steps of V_WMMA_F32_16X16X4_F32.
__global__ __launch_bounds__(256) void gemm_xw_wmma(const float* __restrict__ H,
                                                    const float* __restrict__ W,
                                                    float* __restrict__ XW) {
    const int tile = blockIdx.x * 8 + (threadIdx.x >> 5);
    if (tile >= N_NODES / 16) return;   // wave-uniform: EXEC stays all-1s for WMMA
    const int lane  = threadIdx.x & 31;
    const int l16   = lane & 15;        // M (A frag) / N (B,C,D frags)
    const int hi    = lane >> 4;        // 0: K pair {0,1}/rows 0-7 ; 1: {2,3}/rows 8-15
    const int khalf = hi * 2;
    const size_t rowA = (size_t)(tile * 16 + l16) * HIDDEN;

    v8f acc[8];
#pragma unroll
    for (int nt = 0; nt < 8; ++nt) acc[nt] = (v8f)0.0f;

#pragma unroll 4
    for (int kk = 0; kk < 32; ++kk) {
        const int k = kk * 4 + khalf;
        // A fragment: lane (M=l16) holds H[M][k], H[M][k+1]
        v2f a;
        a.x = H[rowA + k];
        a.y = H[rowA + k + 1];
#pragma unroll
        for (int nt = 0; nt < 8; ++nt) {
            const int n = nt * 16 + l16;
            // B fragment: lane (N=l16) holds W[k][N], W[k+1][N]
            v2f b;
            b.x = W[(size_t)k * HIDDEN + n];
            b.y = W[(size_t)(k + 1) * HIDDEN + n];
            acc[nt] = __builtin_amdgcn_wmma_f32_16x16x4_f32(
                false, a, false, b, (short)0, acc[nt], false, false);
        }
    }

    // D layout: lanes 0-15 -> rows v (VGPR v), lanes 16-31 -> rows v+8; col = l16
#pragma unroll
    for (int v = 0; v < 8; ++v) {
        const int row = tile * 16 + hi * 8 + v;
#pragma unroll
        for (int nt = 0; nt < 8; ++nt) {
            XW[(size_t)row * HIDDEN + nt * 16 + l16] = acc[nt][v];
        }
    }
}

// ------------------------------------------------- edge scatter (message passing)
// 32 lanes per edge; each lane moves 4 consecutive features (coalesced float4
// gather from xw[src], 4 float atomics into agg[dst]).
__global__ void scatter_kernel(const float* __restrict__ xw,
                               const int* __restrict__ src,
                               const int* __restrict__ dst,
                               const float* __restrict__ norm,
                               float* __restrict__ agg) {
    size_t t = (size_t)blockIdx.x * blockDim.x + threadIdx.x;
    int e = (int)(t >> 5);
    if (e >= N_EDGES) return;
    int lane = (int)(t & 31);
    int s = src[e], d = dst[e];
    float c = norm[s] * norm[d];
    const float4 v = *(const float4*)(xw + (size_t)s * HIDDEN + lane * 4);
    float* o = agg + (size_t)d * HIDDEN + lane * 4;
    atomicAdd(o + 0, v.x * c);
    atomicAdd(o + 1, v.y * c);
    atomicAdd(o + 2, v.z * c);
    atomicAdd(o + 3, v.w * c);
}

// --------------------------- fused self-loop + bias + BN(eval) + ReLU, in-place
__global__ void update_kernel(const float* __restrict__ xw,
                              const float* __restrict__ norm,
                              const float* __restrict__ bias,
                              const float* __restrict__ gamma,
                              const float* __restrict__ beta,
                              const float* __restrict__ mean,
                              const float* __restrict__ var,
                              float* __restrict__ agg) {
    size_t t = (size_t)blockIdx.x * blockDim.x + threadIdx.x;
    if (t >= (size_t)N_NODES * 32) return;
    int node = (int)(t >> 5);
    int f0 = (int)(t & 31) * 4;
    float nr = norm[node];
    float sc = nr * nr;               // self_coef = norm^2
    size_t base = (size_t)node * HIDDEN + f0;
#pragma unroll
    for (int j = 0; j < 4; ++j) {
        int f = f0 + j;
        float val = agg[base + j] + xw[base + j] * sc + bias[f];
        val = (val - mean[f]) * (gamma[f] * rsqrtf(var[f] + BN_EPS)) + beta[f];
        agg[base + j] = fmaxf(val, 0.0f);
    }
}

// ------------------------------------------------------------- graph mean pool
__global__ void count_kernel(const int* __restrict__ batch, float* __restrict__ counts) {
    int i = blockIdx.x * blockDim.x + threadIdx.x;
    if (i < N_NODES) atomicAdd(&counts[batch[i]], 1.0f);
}

__global__ void pool_kernel(const float* __restrict__ h, const int* __restrict__ batch,
                            float* __restrict__ sums) {
    size_t t = (size_t)blockIdx.x * blockDim.x + threadIdx.x;
    if (t >= (size_t)N_NODES * 32) return;
    int node = (int)(t >> 5);
    int f0 = (int)(t & 31) * 4;
    int g = batch[node];
    const float4 v = *(const float4*)(h + (size_t)node * HIDDEN + f0);
    float* o = sums + (size_t)g * HIDDEN + f0;
    atomicAdd(o + 0, v.x);
    atomicAdd(o + 1, v.y);
    atomicAdd(o + 2, v.z);
    atomicAdd(o + 3, v.w);
}

// ------------------------------------------------ final tiny GEMM: pooled @ W_out
__global__ void out_kernel(const float* __restrict__ sums, const float* __restrict__ counts,
                           const float* __restrict__ W_out, const float* __restrict__ b_out,
                           float* __restrict__ out) {
    int t = blockIdx.x * blockDim.x + threadIdx.x;
    if (t >= N_GRAPHS * 5) return;
    int g = t / 5, o = t % 5;
    float inv = 1.0f / fmaxf(counts[g], 1.0f);
    float acc = b_out[o];
    for (int f = 0; f < HIDDEN; ++f)
        acc += sums[(size_t)g * HIDDEN + f] * inv * W_out[f * 5 + o];
    out[t] = acc;
}

// ============================================================================
extern "C" void kernel_launch(void* const* d_in, const int* in_sizes, int n_in,
                              void* d_out, int out_size, void* d_ws, size_t ws_size,
                              hipStream_t stream) {
    const float* x        = (const float*)d_in[0];
    const int*   ei       = (const int*)d_in[1];     // [2, N_EDGES]
    const int*   src      = ei;
    const int*   dst      = ei + N_EDGES;
    const int*   batch    = (const int*)d_in[2];
    const float* Ws       = (const float*)d_in[3];   // [4,128,128]
    const float* bs       = (const float*)d_in[4];
    const float* gammas   = (const float*)d_in[5];
    const float* betas    = (const float*)d_in[6];
    const float* run_mean = (const float*)d_in[7];
    const float* run_var  = (const float*)d_in[8];
    const float* W_out    = (const float*)d_in[9];   // [128,5]
    const float* b_out    = (const float*)d_in[10];
    float*       out      = (float*)d_out;

    // workspace layout (floats): 3x node-feature buffers + norm + pool
    const size_t NF = (size_t)N_NODES * HIDDEN;          // 12.8M floats each
    float* bufA   = (float*)d_ws;
    float* bufB   = bufA + NF;                            // xw
    float* bufC   = bufB + NF;
    float* norm   = bufC + NF;                            // N_NODES
    float* sums   = norm + N_NODES;                       // N_GRAPHS*HIDDEN
    float* counts = sums + (size_t)N_GRAPHS * HIDDEN;     // N_GRAPHS (contiguous after sums)

    const int B = 256;
    auto cdiv = [](size_t a, size_t b) { return (int)((a + b - 1) / b); };

    // degree + symmetric normalization
    zero_f32<<<cdiv(N_NODES, B), B, 0, stream>>>(norm, (size_t)N_NODES);
    degree_kernel<<<cdiv(N_EDGES, B), B, 0, stream>>>(dst, norm);
    norm_kernel<<<cdiv(N_NODES, B), B, 0, stream>>>(norm);

    const float* h = x;
    float* aggsel[2] = { bufA, bufC };
    for (int l = 0; l < N_LAYERS; ++l) {
        float* agg = aggsel[l & 1];
        gemm_xw_wmma<<<cdiv(N_NODES / 16, 8), 256, 0, stream>>>(
            h, Ws + (size_t)l * HIDDEN * HIDDEN, bufB);
        zero_f32<<<cdiv(NF, B), B, 0, stream>>>(agg, NF);
        scatter_kernel<<<cdiv((size_t)N_EDGES * 32, B), B, 0, stream>>>(
            bufB, src, dst, norm, agg);
        update_kernel<<<cdiv((size_t)N_NODES * 32, B), B, 0, stream>>>(
            bufB, norm, bs + l * HIDDEN, gammas + l * HIDDEN, betas + l * HIDDEN,
            run_mean + l * HIDDEN, run_var + l * HIDDEN, agg);
        h = agg;
    }

    // pooling (+counts) buffers are contiguous: zero both in one launch
    zero_f32<<<cdiv((size_t)N_GRAPHS * HIDDEN + N_GRAPHS, B), B, 0, stream>>>(
        sums, (size_t)N_GRAPHS * HIDDEN + N_GRAPHS);
    count_kernel<<<cdiv(N_NODES, B), B, 0, stream>>>(batch, counts);
    pool_kernel<<<cdiv((size_t)N_NODES * 32, B), B, 0, stream>>>(h, batch, sums);
    out_kernel<<<cdiv(N_GRAPHS * 5, B), B, 0, stream>>>(sums, counts, W_out, b_out, out);
}